// VectorQuantizer_70403103916654
// MI455X (gfx1250) — compile-verified
//
#include <hip/hip_runtime.h>
#include <math.h>

#define B_ 8
#define S_ 4096
#define SP1 4097
#define D_ 256
#define R_ 512
#define C_ 2048
#define NT_ (C_/16)
#define PPITCH 260   // padded LDS pitch: bank = (4c+d)%64, conflict-free

typedef __attribute__((ext_vector_type(2))) float v2f;
typedef __attribute__((ext_vector_type(8))) float v8f;

// ---- workspace layout (in floats) ----
constexpr size_t OFF_T    = 0;                            // B*R     : t[b,r]
constexpr size_t OFF_PROJ = OFF_T    + (size_t)B_*R_;     // B*D*R   : softmax(cls_proj)
constexpr size_t OFF_EMB  = OFF_PROJ + (size_t)B_*D_*R_;  // B*D*C   : emb_b
constexpr size_t OFF_EMBT = OFF_EMB  + (size_t)B_*D_*C_;  // B*C*D   : emb_b transposed
constexpr size_t OFF_ESQ  = OFF_EMBT + (size_t)B_*C_*D_;  // B*C     : column sq-norms
constexpr size_t OFF_IDX  = OFF_ESQ  + (size_t)B_*C_;     // B*S     : argmin indices (int)
constexpr size_t OFF_HIST = OFF_IDX  + (size_t)B_*S_;     // B*C     : histogram (float)
constexpr size_t OFF_ACC  = OFF_HIST + (size_t)B_*C_;     // 8       : accumulators

// ------------------------------------------------------------------
__global__ void k_zero(float* ws) {
  int i = blockIdx.x * 256 + threadIdx.x;
  if (i < B_*C_)            ws[OFF_HIST + i] = 0.f;
  else if (i < B_*C_ + 8)   ws[OFF_ACC + (i - B_*C_)] = 0.f;
}

// t[b,r] = sum_e cls[b,e] * cls_w[e,r]
__global__ void k_cls_t(const float* __restrict__ x, const float* __restrict__ cls_w,
                        float* ws) {
  __shared__ float cls[D_];
  int b = blockIdx.x >> 1;
  int r = ((blockIdx.x & 1) << 8) + threadIdx.x;
  cls[threadIdx.x] = x[(size_t)b*SP1*D_ + threadIdx.x];
  __syncthreads();
  float s = 0.f;
  for (int e = 0; e < D_; ++e) s += cls[e] * cls_w[(size_t)e*R_ + r];
  ws[OFF_T + (size_t)b*R_ + r] = s;
}

// cls_proj[b,d,r] = softmax_r( cls[b,d]*t[b,r] + cls_b[r] )
__global__ void k_softmax(const float* __restrict__ x, const float* __restrict__ cls_b,
                          float* ws) {
  __shared__ float red[256];
  int b = blockIdx.x >> 8;
  int d = blockIdx.x & 255;
  float cd = x[(size_t)b*SP1*D_ + d];
  const float* t = ws + OFF_T + (size_t)b*R_;
  int r0 = threadIdx.x, r1 = threadIdx.x + 256;
  float l0 = cd * t[r0] + cls_b[r0];
  float l1 = cd * t[r1] + cls_b[r1];
  red[threadIdx.x] = fmaxf(l0, l1); __syncthreads();
  for (int off = 128; off > 0; off >>= 1) {
    if (threadIdx.x < off) red[threadIdx.x] = fmaxf(red[threadIdx.x], red[threadIdx.x + off]);
    __syncthreads();
  }
  float m = red[0]; __syncthreads();
  float e0 = __expf(l0 - m), e1 = __expf(l1 - m);
  red[threadIdx.x] = e0 + e1; __syncthreads();
  for (int off = 128; off > 0; off >>= 1) {
    if (threadIdx.x < off) red[threadIdx.x] += red[threadIdx.x + off];
    __syncthreads();
  }
  float inv = 1.f / red[0];
  float* proj = ws + OFF_PROJ + ((size_t)b*D_ + d)*R_;
  proj[r0] = e0 * inv; proj[r1] = e1 * inv;
}

// emb_b[b] = cls_proj[b] (256x512) @ embeddings (512x2048), fp32 WMMA.
__global__ void k_emb_gemm(const float* __restrict__ emb, float* ws) {
  int w    = blockIdx.x * 4 + (threadIdx.x >> 5);
  int lane = threadIdx.x & 31;
  int b  = w >> 11;
  int mt = (w >> 7) & 15;
  int nt = w & 127;
  const float* A = ws + OFF_PROJ + (size_t)b*D_*R_;
  int mrow  = mt*16 + (lane & 15);
  int ncol  = nt*16 + (lane & 15);
  int khalf = (lane >> 4) << 1;              // lanes 0-15 -> K{0,1}, 16-31 -> K{2,3}
  v8f acc = {};
  for (int k = 0; k < R_; k += 4) {
    v2f a = *(const v2f*)(A + (size_t)mrow*R_ + k + khalf);
    v2f bb;
    bb.x = emb[(size_t)(k + khalf    )*C_ + ncol];
    bb.y = emb[(size_t)(k + khalf + 1)*C_ + ncol];
    acc = __builtin_amdgcn_wmma_f32_16x16x4_f32(false, a, false, bb,
                                                (short)0, acc, false, false);
  }
  float* EB = ws + OFF_EMB  + (size_t)b*D_*C_;
  float* ET = ws + OFF_EMBT + (size_t)b*C_*D_;
  int half = lane >> 4;
#pragma unroll
  for (int i = 0; i < 8; ++i) {              // C layout: vgpr i, lane<16 -> M=i else M=8+i
    int row = mt*16 + half*8 + i;
    float v = acc[i];
    EB[(size_t)row*C_ + ncol] = v;
    ET[(size_t)ncol*D_ + row] = v;
  }
}

// e_sq[b,c] = sum_d emb_b[b,d,c]^2
__global__ void k_esq(float* ws) {
  int g = blockIdx.x * 256 + threadIdx.x;
  int b = g / C_, c = g % C_;
  const float* EB = ws + OFF_EMB + (size_t)b*D_*C_;
  float s = 0.f;
  for (int d = 0; d < D_; ++d) { float v = EB[(size_t)d*C_ + c]; s += v*v; }
  ws[OFF_ESQ + g] = s;
}

// indices[b,s] = argmin_c ( e_sq[b,c] - 2 * dot(pos[b,s,:], emb_b[b,:,c]) )
// One wave per 16-row s-tile. c-panels are async-copied (transposed scatter) into a
// double-buffered padded LDS panel; copy of panel ct+1 overlaps WMMA on panel ct.
__global__ void k_argmin(const float* __restrict__ x, float* ws) {
  __shared__ float panel[2 * 16 * PPITCH];   // 33,280 B
  int lane = threadIdx.x & 31;
  int wid  = threadIdx.x >> 5;
  int b  = blockIdx.x >> 5;                  // 32 blocks/batch * 8 waves = 256 s-tiles
  int st = ((blockIdx.x & 31) << 3) + wid;
  const float* pos = x + ((size_t)b*SP1 + 1)*D_;
  const float* EB  = ws + OFF_EMB + (size_t)b*D_*C_;
  const float* esq = ws + OFF_ESQ + (size_t)b*C_;
  int mrow  = st*16 + (lane & 15);
  int khalf = (lane >> 4) << 1;
  const unsigned lds_base = (unsigned)(uintptr_t)panel;  // low 32 bits = LDS byte offset

  float bestv[8]; int besti[8];
#pragma unroll
  for (int i = 0; i < 8; ++i) { bestv[i] = 3.4e38f; besti[i] = 0; }

  const float* prow = pos + (size_t)mrow*D_;

  // issue async copy of panel 0 into buffer 0 (16 async b32 per wave)
#pragma unroll 4
  for (int j = 0; j < 16; ++j) {
    int e = j*256 + threadIdx.x;           // 0..4095
    int d = e >> 4, c = e & 15;
    const float* g = EB + (size_t)d*C_ + c;
    unsigned loff = lds_base + (unsigned)((c*PPITCH + d) * 4);
    asm volatile("global_load_async_to_lds_b32 %0, %1, off"
                 :: "v"(loff), "v"(g) : "memory");
  }

  for (int ct = 0; ct < NT_; ++ct) {
    int cur = ct & 1;
    if (ct + 1 < NT_) {
      int nb = cur ^ 1;
#pragma unroll 4
      for (int j = 0; j < 16; ++j) {       // prefetch next panel into other buffer
        int e = j*256 + threadIdx.x;
        int d = e >> 4, c = e & 15;
        const float* g = EB + (size_t)d*C_ + (ct + 1)*16 + c;
        unsigned loff = lds_base + (unsigned)(((nb*16 + c)*PPITCH + d) * 4);
        asm volatile("global_load_async_to_lds_b32 %0, %1, off"
                     :: "v"(loff), "v"(g) : "memory");
      }
      // 32 outstanding; first 16 (= panel ct) complete when count <= 16
      asm volatile("s_wait_asynccnt 0x10" ::: "memory");
    } else {
      asm volatile("s_wait_asynccnt 0x0" ::: "memory");
    }
    __syncthreads();                        // all waves' slices of panel ct visible

    v8f acc = {};
    const float* P = panel + (cur*16 + (lane & 15)) * PPITCH;
    for (int k = 0; k < D_; k += 4) {
      v2f a  = *(const v2f*)(prow + k + khalf);   // global b64, L0-resident
      v2f bb = *(const v2f*)(P    + k + khalf);   // single ds_load_b64, conflict-free
      acc = __builtin_amdgcn_wmma_f32_16x16x4_f32(false, a, false, bb,
                                                  (short)0, acc, false, false);
    }
    int   cidx = ct*16 + (lane & 15);
    float es   = esq[cidx];
#pragma unroll
    for (int i = 0; i < 8; ++i) {
      float val = es - 2.f*acc[i];
      if (val < bestv[i]) { bestv[i] = val; besti[i] = cidx; }  // strict < keeps earliest
    }
    __syncthreads();                        // done reading buf[cur] before it is re-filled
  }

  // reduce across the 16 lanes of each half (masks <=8 stay within halves)
#pragma unroll
  for (int i = 0; i < 8; ++i) {
    float v = bestv[i]; int ix = besti[i];
    for (int m = 8; m > 0; m >>= 1) {
      float ov = __shfl_xor(v, m, 32);
      int   oi = __shfl_xor(ix, m, 32);
      if (ov < v || (ov == v && oi < ix)) { v = ov; ix = oi; }
    }
    if ((lane & 15) == i) {
      int srow = st*16 + (lane >> 4)*8 + i;
      ((int*)(ws + OFF_IDX))[(size_t)b*S_ + srow] = ix;
    }
  }
}

// quantized_out + squared-error accumulation + histogram
__global__ void k_gather(const float* __restrict__ x, float* __restrict__ out, float* ws) {
  __shared__ float red[256];
  long blk = blockIdx.x;
  if (blk >= (long)B_*S_) {                  // cls-token passthrough rows
    int b = (int)(blk - (long)B_*S_);
    out[((size_t)b*SP1)*D_ + threadIdx.x] = x[((size_t)b*SP1)*D_ + threadIdx.x];
    return;
  }
  int b = (int)(blk >> 12);
  int s = (int)(blk & 4095);
  int idx = ((const int*)(ws + OFF_IDX))[(size_t)b*S_ + s];
  const float* q = ws + OFF_EMBT + ((size_t)b*C_ + idx)*D_;
  const float* p = x + ((size_t)b*SP1 + 1 + s)*D_;
  float qv = q[threadIdx.x];
  float pv = p[threadIdx.x];
  out[((size_t)b*SP1 + 1 + s)*D_ + threadIdx.x] = qv;
  float dd = qv - pv;
  red[threadIdx.x] = dd*dd;
  __syncthreads();
  for (int off = 128; off > 0; off >>= 1) {
    if (threadIdx.x < off) red[threadIdx.x] += red[threadIdx.x + off];
    __syncthreads();
  }
  if (threadIdx.x == 0) {
    atomicAdd(ws + OFF_ACC, red[0]);
    atomicAdd(ws + OFF_HIST + (size_t)b*C_ + idx, 1.0f);
  }
}

__global__ void k_final(float* ws, float* out) {
  __shared__ float red[256];
  float psum = 0.f;
  for (int b = 0; b < B_; ++b) {
    const float* h = ws + OFF_HIST + (size_t)b*C_;
    float ent = 0.f;
    for (int c = threadIdx.x; c < C_; c += 256) {
      float p = h[c] * (1.0f / S_);
      ent += p * __logf(p + 1e-10f);
    }
    red[threadIdx.x] = ent; __syncthreads();
    for (int off = 128; off > 0; off >>= 1) {
      if (threadIdx.x < off) red[threadIdx.x] += red[threadIdx.x + off];
      __syncthreads();
    }
    if (threadIdx.x == 0) psum += __expf(-red[0]);
    __syncthreads();
  }
  if (threadIdx.x == 0) {
    float e = ws[OFF_ACC] / (float)((size_t)B_*S_*D_);
    size_t NOUT = (size_t)B_*SP1*D_;
    out[NOUT + 0] = psum / (float)B_;  // perplexity
    out[NOUT + 1] = e;                 // e_latent_loss
    out[NOUT + 2] = e;                 // q_latent_loss (same forward value)
    out[NOUT + 3] = e * 0.25f + e;     // vq_loss
  }
}

extern "C" void kernel_launch(void* const* d_in, const int* in_sizes, int n_in,
                              void* d_out, int out_size, void* d_ws, size_t ws_size,
                              hipStream_t stream) {
  const float* x          = (const float*)d_in[0];
  const float* embeddings = (const float*)d_in[1];
  const float* cls_w      = (const float*)d_in[2];
  const float* cls_b      = (const float*)d_in[3];
  float* out = (float*)d_out;
  float* ws  = (float*)d_ws;

  k_zero    <<<(B_*C_ + 8 + 255)/256, 256, 0, stream>>>(ws);
  k_cls_t   <<<B_*2,              256, 0, stream>>>(x, cls_w, ws);
  k_softmax <<<B_*D_,             256, 0, stream>>>(x, cls_b, ws);
  k_emb_gemm<<<4096,              128, 0, stream>>>(embeddings, ws);
  k_esq     <<<(B_*C_)/256,       256, 0, stream>>>(ws);
  k_argmin  <<<256,               256, 0, stream>>>(x, ws);
  k_gather  <<<B_*S_ + B_,        256, 0, stream>>>(x, out, ws);
  k_final   <<<1,                 256, 0, stream>>>(ws, out);
}